// SemiTemporalSwapGNN_28071906246937
// MI455X (gfx1250) — compile-verified
//
#include <hip/hip_runtime.h>
#include <math.h>

#define NN 100000
#define EE 1600000
#define TT 4
#define NLK 15

typedef __attribute__((ext_vector_type(2))) float v2f;
typedef __attribute__((ext_vector_type(8))) float v8f;

// ---------------- stats: mean/std of requests[NL:] (ddof=1) ----------------

__global__ void zero_stats_kernel(double* stats) {
    if (threadIdx.x == 0) { stats[0] = 0.0; stats[1] = 0.0; stats[2] = 0.0; stats[3] = 0.0; }
}

__global__ __launch_bounds__(256)
void reduce_stats_kernel(const float* __restrict__ req, double* __restrict__ stats, int n) {
    __shared__ double s1[256];
    __shared__ double s2[256];
    int tid = threadIdx.x;
    int i = blockIdx.x * blockDim.x + tid;
    int stride = gridDim.x * blockDim.x;
    double a = 0.0, b = 0.0;
    int tail = n - NLK;
    for (int k = i; k < tail; k += stride) {
        double v = (double)req[NLK + k];
        a += v; b += v * v;
    }
    s1[tid] = a; s2[tid] = b;
    __syncthreads();
    for (int s = 128; s > 0; s >>= 1) {
        if (tid < s) { s1[tid] += s1[tid + s]; s2[tid] += s2[tid + s]; }
        __syncthreads();
    }
    if (tid == 0) { atomicAdd(&stats[0], s1[0]); atomicAdd(&stats[1], s2[0]); }
}

__global__ void finalize_stats_kernel(double* stats, int n) {
    if (threadIdx.x == 0) {
        double cnt = (double)(n - NLK);
        double s = stats[0], ss = stats[1];
        double mean = s / cnt;
        double var = (ss - s * s / cnt) / (cnt - 1.0);
        stats[2] = mean;
        stats[3] = 1.0 / sqrt(var);
    }
}

// ---------------- build layer-0 features (N x 16, zero padded) ----------------

__global__ __launch_bounds__(256)
void build_x0_kernel(const int* __restrict__ nt, const float* __restrict__ req,
                     const float* __restrict__ ti, const float* __restrict__ emb,
                     const double* __restrict__ stats, float* __restrict__ x, int n) {
    int i = blockIdx.x * blockDim.x + threadIdx.x;
    if (i >= n) return;
    float mean = (float)stats[2];
    float istd = (float)stats[3];
    int t = nt[i];
    float r = req[i];
    float rf = (i < NLK) ? r : (r - mean) * istd;
    float* xi = x + (size_t)i * 16;
    xi[0] = emb[t * 3 + 0];
    xi[1] = emb[t * 3 + 1];
    xi[2] = emb[t * 3 + 2];
    xi[3] = rf;
    xi[4] = ti[i];
#pragma unroll
    for (int c = 5; c < 16; ++c) xi[c] = 0.f;
}

// ---------------- fused GEMM + attention projections via V_WMMA_F32_16X16X4_F32 ----
// B tile (16x16 in LDS, zero padded):
//   cols 0..11  : W[k][n]
//   col 12      : sum_c W[k][0*6+c]*a_s[0*6+c]   -> e_s head 0
//   col 13      : sum_c W[k][1*6+c]*a_s[1*6+c]   -> e_s head 1
//   col 14/15   : same with a_d                   -> e_d head 0/1
// One wave per 16-row tile of x; D cols 0..11 -> h, cols 12..15 -> es/ed.

__global__ __launch_bounds__(256)
void gemm_wmma_kernel(const float* __restrict__ x, const float* __restrict__ W,
                      const float* __restrict__ a_s, const float* __restrict__ a_d,
                      float* __restrict__ h, float* __restrict__ es, float* __restrict__ ed,
                      int nTiles, int din) {
    __shared__ float Wl[16 * 16];
    int tid = threadIdx.x;
    {
        int k = tid >> 4;
        int n = tid & 15;
        float v = 0.f;
        if (k < din) {
            if (n < 12) {
                v = W[k * 12 + n];
            } else {
                const float* av = (n < 14) ? a_s : a_d;
                int hd = n & 1;
                float accv = 0.f;
#pragma unroll
                for (int c = 0; c < 6; ++c) accv += W[k * 12 + hd * 6 + c] * av[hd * 6 + c];
                v = accv;
            }
        }
        Wl[k * 16 + n] = v;
    }
    __syncthreads();

    int wave = (int)((blockIdx.x * blockDim.x + tid) >> 5);
    if (wave >= nTiles) return;              // wave-uniform: EXEC all-1s at WMMA
    int lane = tid & 31;
    int nlo = lane & 15;
    int khalf = lane >> 4;
    int rowBase = wave * 16;

    v8f c = {};
    int kSteps = (din + 3) >> 2;
    const float* xr = x + (size_t)(rowBase + nlo) * 16;
    for (int kk = 0; kk < kSteps; ++kk) {
        int k0 = kk * 4 + khalf * 2;         // lanes 0-15: K=k0,k0+1 ; lanes 16-31: +2
        v2f a, b;
        a.x = xr[k0];                         // x zero-padded beyond din
        a.y = xr[k0 + 1];
        b.x = Wl[k0 * 16 + nlo];
        b.y = Wl[(k0 + 1) * 16 + nlo];
#if defined(__HIP_DEVICE_COMPILE__)
        c = __builtin_amdgcn_wmma_f32_16x16x4_f32(
                /*neg_a=*/false, a, /*neg_b=*/false, b,
                /*c_mod=*/(short)0, c, /*reuse_a=*/false, /*reuse_b=*/false);
#endif
    }
    // D: VGPR r holds (M = r + 8*khalf, N = nlo)
    if (nlo < 12) {
#pragma unroll
        for (int r = 0; r < 8; ++r) {
            h[(size_t)(rowBase + r + 8 * khalf) * 12 + nlo] = c[r];
        }
    } else {
        int hd = nlo & 1;
        float* dstp = (nlo < 14) ? es : ed;
#pragma unroll
        for (int r = 0; r < 8; ++r) {
            dstp[(size_t)(rowBase + r + 8 * khalf) * 2 + hd] = c[r];
        }
    }
}

// ---------------- per-layer workspace init ----------------

__global__ __launch_bounds__(256)
void init_layer_kernel(float* __restrict__ m, float* __restrict__ den,
                       float* __restrict__ acc, int n) {
    int i = blockIdx.x * blockDim.x + threadIdx.x;
    if (i >= n) return;
    float2 ninf = make_float2(-INFINITY, -INFINITY);
    *(float2*)(m + (size_t)i * 2) = ninf;
    *(float2*)(den + (size_t)i * 2) = make_float2(0.f, 0.f);
    float4 z = make_float4(0.f, 0.f, 0.f, 0.f);
    float4* a4 = (float4*)(acc + (size_t)i * 12);
    a4[0] = z; a4[1] = z; a4[2] = z;
}

// ---------------- edge sweeps ----------------

__device__ __forceinline__ void atomicMaxFloat(float* addr, float val) {
    if (val >= 0.f) atomicMax((int*)addr, __float_as_int(val));
    else            atomicMin((unsigned int*)addr, __float_as_uint(val));
}

__device__ __forceinline__ void edge_sd(const int* ei0, const int* ei1,
                                        int e, int E_, int& s, int& d) {
    if (e < E_) { s = ei0[e]; d = ei1[e]; }
    else        { s = e - E_; d = e - E_; }
}

__device__ __forceinline__ float lrelu(float v) { return v > 0.f ? v : 0.2f * v; }

__global__ __launch_bounds__(256)
void edge_max_kernel(const int* __restrict__ ei0, const int* __restrict__ ei1,
                     const float* __restrict__ es, const float* __restrict__ ed,
                     float* __restrict__ m, int E_, int NE) {
    int e = blockIdx.x * blockDim.x + threadIdx.x;
    if (e >= NE) return;
    int s, d; edge_sd(ei0, ei1, e, E_, s, d);
    float2 ess = *(const float2*)(es + (size_t)s * 2);
    float2 edd = *(const float2*)(ed + (size_t)d * 2);
    atomicMaxFloat(&m[d * 2 + 0], lrelu(ess.x + edd.x));
    atomicMaxFloat(&m[d * 2 + 1], lrelu(ess.y + edd.y));
}

__global__ __launch_bounds__(256)
void fix_m_kernel(float* __restrict__ m, int n2) {
    int i = blockIdx.x * blockDim.x + threadIdx.x;
    if (i >= n2) return;
    float v = m[i];
    m[i] = isfinite(v) ? v : 0.0f;
}

__global__ __launch_bounds__(256)
void edge_denom_kernel(const int* __restrict__ ei0, const int* __restrict__ ei1,
                       const float* __restrict__ es, const float* __restrict__ ed,
                       const float* __restrict__ m, float* __restrict__ den,
                       int E_, int NE) {
    int e = blockIdx.x * blockDim.x + threadIdx.x;
    if (e >= NE) return;
    int s, d; edge_sd(ei0, ei1, e, E_, s, d);
    float2 ess = *(const float2*)(es + (size_t)s * 2);
    float2 edd = *(const float2*)(ed + (size_t)d * 2);
    float2 md  = *(const float2*)(m + (size_t)d * 2);
    atomicAdd(&den[d * 2 + 0], __expf(lrelu(ess.x + edd.x) - md.x));
    atomicAdd(&den[d * 2 + 1], __expf(lrelu(ess.y + edd.y) - md.y));
}

__global__ __launch_bounds__(256)
void edge_aggr_kernel(const int* __restrict__ ei0, const int* __restrict__ ei1,
                      const float* __restrict__ es, const float* __restrict__ ed,
                      const float* __restrict__ m, const float* __restrict__ den,
                      const float* __restrict__ h, float* __restrict__ acc,
                      int E_, int NE) {
    int e = blockIdx.x * blockDim.x + threadIdx.x;
    if (e >= NE) return;
    int s, d; edge_sd(ei0, ei1, e, E_, s, d);
    float2 ess = *(const float2*)(es + (size_t)s * 2);
    float2 edd = *(const float2*)(ed + (size_t)d * 2);
    float2 md  = *(const float2*)(m + (size_t)d * 2);
    float2 dd  = *(const float2*)(den + (size_t)d * 2);
    float a0 = __expf(lrelu(ess.x + edd.x) - md.x) / (dd.x + 1e-16f);
    float a1 = __expf(lrelu(ess.y + edd.y) - md.y) / (dd.y + 1e-16f);
    const float4* hs4 = (const float4*)(h + (size_t)s * 12);   // 48B rows, 16B aligned
    float4 h0 = hs4[0], h1 = hs4[1], h2 = hs4[2];
    float* ao = acc + (size_t)d * 12;
    atomicAdd(&ao[0],  a0 * h0.x);
    atomicAdd(&ao[1],  a0 * h0.y);
    atomicAdd(&ao[2],  a0 * h0.z);
    atomicAdd(&ao[3],  a0 * h0.w);
    atomicAdd(&ao[4],  a0 * h1.x);
    atomicAdd(&ao[5],  a0 * h1.y);
    atomicAdd(&ao[6],  a1 * h1.z);
    atomicAdd(&ao[7],  a1 * h1.w);
    atomicAdd(&ao[8],  a1 * h2.x);
    atomicAdd(&ao[9],  a1 * h2.y);
    atomicAdd(&ao[10], a1 * h2.z);
    atomicAdd(&ao[11], a1 * h2.w);
}

// ---------------- bias (+ optional relu) and write to next-x / output ----------------

__global__ __launch_bounds__(256)
void finalize_kernel(const float* __restrict__ acc, const float* __restrict__ b,
                     float* __restrict__ out, int strideOut, int doRelu, int n) {
    int idx = blockIdx.x * blockDim.x + threadIdx.x;
    if (idx >= n * 12) return;
    int i = idx / 12;
    int ch = idx - i * 12;
    float v = acc[idx] + b[ch];
    if (doRelu) v = fmaxf(v, 0.f);
    out[(size_t)i * strideOut + ch] = v;
}

// ---------------- host orchestration ----------------

extern "C" void kernel_launch(void* const* d_in, const int* in_sizes, int n_in,
                              void* d_out, int out_size, void* d_ws, size_t ws_size,
                              hipStream_t stream) {
    (void)in_sizes; (void)n_in; (void)out_size; (void)ws_size;
    const int N_ = NN, E_ = EE;
    const int NE = E_ + N_;

    const int*   node_types = (const int*)d_in[0];
    const float* requests   = (const float*)d_in[1];
    const float* time_index = (const float*)d_in[2];
    const int*   edge_index = (const int*)d_in[3];
    const float* emb        = (const float*)d_in[4];

    double* stats = (double*)d_ws;
    float* fbase = (float*)((char*)d_ws + 64);
    float* x   = fbase;                   // N*16 (padded)
    float* h   = x   + (size_t)N_ * 16;   // N*12 (16B-aligned rows)
    float* es  = h   + (size_t)N_ * 12;   // N*2
    float* ed  = es  + (size_t)N_ * 2;    // N*2
    float* mW  = ed  + (size_t)N_ * 2;    // N*2
    float* den = mW  + (size_t)N_ * 2;    // N*2
    float* acc = den + (size_t)N_ * 2;    // N*12

    const int grN   = (N_ + 255) / 256;
    const int gr2N  = (2 * N_ + 255) / 256;
    const int gr12N = (12 * N_ + 255) / 256;
    const int grE   = (NE + 255) / 256;
    const int nTiles = N_ / 16;                        // 6250
    const int grTiles = (nTiles + 7) / 8;              // 8 waves / 256-thread block

    for (int t = 0; t < TT; ++t) {
        const int*   nt  = node_types + (size_t)t * N_;
        const float* req = requests   + (size_t)t * N_;
        const float* ti  = time_index + (size_t)t * N_;
        const int*   ei0 = edge_index + (size_t)t * 2 * E_;
        const int*   ei1 = ei0 + E_;
        float* outT = (float*)d_out + (size_t)t * N_ * 12;

        zero_stats_kernel<<<1, 32, 0, stream>>>(stats);
        reduce_stats_kernel<<<256, 256, 0, stream>>>(req, stats, N_);
        finalize_stats_kernel<<<1, 32, 0, stream>>>(stats, N_);
        build_x0_kernel<<<grN, 256, 0, stream>>>(nt, req, ti, emb, stats, x, N_);

        for (int l = 0; l < 4; ++l) {
            const float* W   = (const float*)d_in[5 + 4 * l];
            const float* as_ = (const float*)d_in[6 + 4 * l];
            const float* ad_ = (const float*)d_in[7 + 4 * l];
            const float* b_  = (const float*)d_in[8 + 4 * l];
            int din = (l == 0) ? 5 : 12;

            gemm_wmma_kernel<<<grTiles, 256, 0, stream>>>(x, W, as_, ad_, h, es, ed,
                                                          nTiles, din);
            init_layer_kernel<<<grN, 256, 0, stream>>>(mW, den, acc, N_);
            edge_max_kernel<<<grE, 256, 0, stream>>>(ei0, ei1, es, ed, mW, E_, NE);
            fix_m_kernel<<<gr2N, 256, 0, stream>>>(mW, 2 * N_);
            edge_denom_kernel<<<grE, 256, 0, stream>>>(ei0, ei1, es, ed, mW, den, E_, NE);
            edge_aggr_kernel<<<grE, 256, 0, stream>>>(ei0, ei1, es, ed, mW, den, h, acc, E_, NE);
            if (l < 3) {
                finalize_kernel<<<gr12N, 256, 0, stream>>>(acc, b_, x, 16, 1, N_);
            } else {
                finalize_kernel<<<gr12N, 256, 0, stream>>>(acc, b_, outT, 12, 0, N_);
            }
        }
    }
}